// FineMatching_90838558310960
// MI455X (gfx1250) — compile-verified
//
#include <hip/hip_runtime.h>
#include <math.h>

typedef __attribute__((ext_vector_type(2))) float v2f;
typedef __attribute__((ext_vector_type(8))) float v8f;
typedef __attribute__((ext_vector_type(4))) int   v4i;

#define AS1 __attribute__((address_space(1)))
#define AS3 __attribute__((address_space(3)))

#define WWIN 25          // 5x5 window
#define CCH  128         // channels
#define ROW_STRIDE 132   // 128 + 4 pad dwords: rows 16B aligned, conflict-free b64 feeds
#define F1_DWORDS (WWIN * ROW_STRIDE)   // 25 data rows only (3300 dwords)
#define F0_OFF    F1_DWORDS             // 3300 (byte 13200, 16B aligned)
#define SIM_OFF   (F0_OFF + CCH)        // 3428
#define LDS_PER_WAVE (SIM_OFF + 32)     // 3460 dwords = 13840 B per wave
#define WPB 2                           // waves (rows m) per block

#if __has_builtin(__builtin_amdgcn_global_load_async_to_lds_b128)
#define HAVE_ASYNC_LDS 1
#else
#define HAVE_ASYNC_LDS 0
#endif

__device__ __forceinline__ void wait_asynccnt0() {
    asm volatile("s_wait_asynccnt 0" ::: "memory");
}

__global__ __launch_bounds__(WPB * 32)
void fine_matching_kernel(const float* __restrict__ feat_f0,
                          const float* __restrict__ feat_f1,
                          const int*   __restrict__ zs_i,
                          const float* __restrict__ zs_f,
                          float* __restrict__ out,
                          int M, int Ng)
{
    __shared__ float lds[WPB * LDS_PER_WAVE];
    const int lane = threadIdx.x & 31;
    const int wid  = threadIdx.x >> 5;
    const int m    = blockIdx.x * WPB + wid;
    if (m >= M) return;

    float* F1  = lds + wid * LDS_PER_WAVE;
    float* F0v = F1 + F0_OFF;
    float* SIM = F1 + SIM_OFF;

    // ---- stream feat_f1[m] (25x128 f32, 12.8 KB) into LDS ----
    const float4* f1g = (const float4*)(feat_f1 + (size_t)m * (WWIN * CCH));
#if HAVE_ASYNC_LDS
    // CDNA5 async copy: global -> LDS with no VGPR staging; tracked by ASYNCcnt.
    #pragma unroll 5
    for (int r = 0; r < WWIN; ++r) {
        AS1 v4i* gp = (AS1 v4i*)(f1g + r * 32 + lane);              // addrspacecast + const strip
        AS3 v4i* lp = (AS3 v4i*)(F1 + r * ROW_STRIDE + lane * 4);   // addrspacecast generic->LDS
        __builtin_amdgcn_global_load_async_to_lds_b128(gp, lp, 0, 0);
    }
#else
    #pragma unroll 5
    for (int r = 0; r < WWIN; ++r) {
        float4 v = f1g[r * 32 + lane];
        *(float4*)(F1 + r * ROW_STRIDE + lane * 4) = v;
    }
#endif

    // ---- build f0 query vector (overlaps with the async stream) ----
    float4 f0;
    if (m < Ng) {
        f0 = *(const float4*)(feat_f0 + ((size_t)m * WWIN + WWIN / 2) * CCH + lane * 4);
    } else {
        const int mz = m - Ng;
        // radius = W//2 = 2 ; ix = (grid_x + 1) * (W-1)/2 = (grid_x + 1) * 2
        const float gx = (zs_f[mz * 2 + 0] - (float)zs_i[mz * 2 + 0]) * 0.5f;
        const float gy = (zs_f[mz * 2 + 1] - (float)zs_i[mz * 2 + 1]) * 0.5f;
        const float ix = (gx + 1.0f) * 2.0f;
        const float iy = (gy + 1.0f) * 2.0f;
        const float ix0 = fminf(fmaxf(floorf(ix), 0.0f), 4.0f);
        const float iy0 = fminf(fmaxf(floorf(iy), 0.0f), 4.0f);
        const float ix1 = fminf(ix0 + 1.0f, 4.0f);
        const float iy1 = fminf(iy0 + 1.0f, 4.0f);
        const float wx1 = ix - ix0, wx0 = 1.0f - wx1;
        const float wy1 = iy - iy0, wy0 = 1.0f - wy1;
        const int i00 = (int)iy0 * 5 + (int)ix0;
        const int i01 = (int)iy0 * 5 + (int)ix1;
        const int i10 = (int)iy1 * 5 + (int)ix0;
        const int i11 = (int)iy1 * 5 + (int)ix1;
        const float w00 = wy0 * wx0, w01 = wy0 * wx1, w10 = wy1 * wx0, w11 = wy1 * wx1;
        const float* base = feat_f0 + (size_t)m * (WWIN * CCH) + lane * 4;
        float4 a = *(const float4*)(base + i00 * CCH);
        float4 b = *(const float4*)(base + i01 * CCH);
        float4 c = *(const float4*)(base + i10 * CCH);
        float4 d = *(const float4*)(base + i11 * CCH);
        f0.x = w00 * a.x + w01 * b.x + w10 * c.x + w11 * d.x;
        f0.y = w00 * a.y + w01 * b.y + w10 * c.y + w11 * d.y;
        f0.z = w00 * a.z + w01 * b.z + w10 * c.z + w11 * d.z;
        f0.w = w00 * a.w + w01 * b.w + w10 * c.w + w11 * d.w;
    }
    *(float4*)(F0v + lane * 4) = f0;

#if HAVE_ASYNC_LDS
    wait_asynccnt0();   // feat_f1 tile resident in LDS before WMMA feeds
#endif

    // ---- sim = F1(25x128) x f0(128) via f32 WMMA 16x16x4, two 16-row M tiles ----
    // A layout  (16x4 f32): lanes 0-15 -> M=lane, {K0,K1}; lanes 16-31 -> M=lane-16, {K2,K3}
    // B layout  ( 4x16 f32): lanes 0-15 -> N=lane, {K0,K1}; lanes 16-31 -> N=lane-16, {K2,K3}
    // B columns all replicate f0 (N redundancy is free: kernel is HBM bound).
    // Tile-2 rows 25..31 read junk/zero LDS; they only pollute ignored D rows.
    const int half = lane >> 4;
    const int l16  = lane & 15;
    const int koff = 2 * half;
    v8f acc0 = {};
    v8f acc1 = {};
    #pragma unroll 8
    for (int kb = 0; kb < CCH; kb += 4) {
        v2f bv = *(const v2f*)(F0v + kb + koff);
        v2f a0 = *(const v2f*)(F1 + l16 * ROW_STRIDE + kb + koff);
        v2f a1 = *(const v2f*)(F1 + (16 + l16) * ROW_STRIDE + kb + koff);
        acc0 = __builtin_amdgcn_wmma_f32_16x16x4_f32(false, a0, false, bv,
                                                     (short)0, acc0, false, false);
        acc1 = __builtin_amdgcn_wmma_f32_16x16x4_f32(false, a1, false, bv,
                                                     (short)0, acc1, false, false);
    }

    // ---- scatter the 25 sims (D: VGPR v, lanes 0-15 -> M=v, lanes 16-31 -> M=v+8) ----
    if (lane == 0) {
        #pragma unroll
        for (int v = 0; v < 8; ++v) { SIM[v] = acc0[v]; SIM[16 + v] = acc1[v]; }
    } else if (lane == 16) {
        #pragma unroll
        for (int v = 0; v < 8; ++v) { SIM[8 + v] = acc0[v]; }
        SIM[24] = acc1[0];
    }

    // ---- softmax over 25 + spatial moments (lane-per-r, shfl reductions) ----
    const float temp = 0.08838834764831845f; // 1/sqrt(128)
    float x = (lane < WWIN) ? SIM[lane] * temp : -1e30f;
    float mx = x;
    #pragma unroll
    for (int off = 16; off; off >>= 1) mx = fmaxf(mx, __shfl_xor(mx, off));
    float e = (lane < WWIN) ? expf(x - mx) : 0.0f;
    float se = e;
    #pragma unroll
    for (int off = 16; off; off >>= 1) se += __shfl_xor(se, off);
    const float h = e / se;

    const int ci = lane % 5, ri = lane / 5;
    const float gxn = -1.0f + 0.5f * (float)ci;   // linspace(-1,1,5)
    const float gyn = -1.0f + 0.5f * (float)ri;
    float cx = h * gxn, cy = h * gyn;
    float sx2 = h * gxn * gxn, sy2 = h * gyn * gyn;
    #pragma unroll
    for (int off = 16; off; off >>= 1) {
        cx  += __shfl_xor(cx, off);
        cy  += __shfl_xor(cy, off);
        sx2 += __shfl_xor(sx2, off);
        sy2 += __shfl_xor(sy2, off);
    }
    const float vx = sx2 - cx * cx;
    const float vy = sy2 - cy * cy;
    const float stdv = sqrtf(fmaxf(vx, 1e-10f)) + sqrtf(fmaxf(vy, 1e-10f));

    if (lane == 0) {
        out[(size_t)m * 3 + 0] = cx;
        out[(size_t)m * 3 + 1] = cy;
        out[(size_t)m * 3 + 2] = stdv;
    }
}

extern "C" void kernel_launch(void* const* d_in, const int* in_sizes, int n_in,
                              void* d_out, int out_size, void* d_ws, size_t ws_size,
                              hipStream_t stream) {
    const float* feat_f0 = (const float*)d_in[0];
    const float* feat_f1 = (const float*)d_in[1];
    const int*   zs_i    = (const int*)d_in[2];
    const float* zs_f    = (const float*)d_in[3];
    // d_in[4] is W (== 5); layout constants are compiled in.
    const int M  = in_sizes[0] / (WWIN * CCH);
    const int Nz = in_sizes[2] / 2;
    const int Ng = M - Nz;
    float* out = (float*)d_out;

    const int blocks = (M + WPB - 1) / WPB;
    fine_matching_kernel<<<blocks, WPB * 32, 0, stream>>>(feat_f0, feat_f1,
                                                          zs_i, zs_f, out, M, Ng);
}